// PoissonGaussianReadout_84782654423082
// MI455X (gfx1250) — compile-verified
//
#include <hip/hip_runtime.h>
#include <math.h>

typedef __attribute__((ext_vector_type(2))) float v2f;
typedef __attribute__((ext_vector_type(8))) float v8f;

#define B_  32
#define C_  256
#define H_  64
#define WD_ 64
#define P_  (H_ * WD_)
#define N_  8192

// ---------------------------------------------------------------------------
// Kernel 1: tile-transpose x (B,C,H,W) -> xt (B,H,W,C) so channel vectors are
// contiguous (1KB rows). Coalesced reads over pixels, coalesced writes over
// channels, 32x33 LDS tile to kill bank conflicts.
// ---------------------------------------------------------------------------
__global__ __launch_bounds__(256) void transpose_x(const float* __restrict__ x,
                                                   float* __restrict__ xt) {
  __shared__ float tile[32][33];
  const int b  = blockIdx.z;
  const int c0 = blockIdx.y * 32;
  const int p0 = blockIdx.x * 32;
  const float* xb  = x  + (size_t)b * C_ * P_;
  float*       xtb = xt + (size_t)b * P_ * C_;

  const int tp = threadIdx.x & 31;
  const int tc = threadIdx.x >> 5;
#pragma unroll
  for (int i = 0; i < 4; ++i) {
    const int c = tc + i * 8;
    tile[c][tp] = xb[(size_t)(c0 + c) * P_ + (p0 + tp)];
  }
  __syncthreads();
  const int wc = threadIdx.x & 31;
  const int wp = threadIdx.x >> 5;
#pragma unroll
  for (int i = 0; i < 4; ++i) {
    const int p = wp + i * 8;
    xtb[(size_t)(p0 + p) * C_ + (c0 + wc)] = tile[wc][p];
  }
}

// ---------------------------------------------------------------------------
// Kernel 2: per 16-n tile: z[b, n0+j] = diag( Wtile(16xC) x S_b(Cx16) ),
// accumulated with V_WMMA_F32_16X16X4_F32 in K=4 channel chunks.
//   A (16x4): lane m = n, rows from the LDS W tile (ds b64)
//   B (4x16): lane col = n, bilinear 4-corner combine of gathered x'
// Each wave runs TWO batches per pass (shared A operand, 2x MLP), 2 passes.
// PS/CS = pixel/channel strides (elements): <C_,1> transposed, <1,P_> raw.
// ---------------------------------------------------------------------------
template <int PS, int CS>
__global__ __launch_bounds__(256) void readout_wmma(
    const float* __restrict__ xsrc, const float* __restrict__ mu,
    const float* __restrict__ Wm, const float* __restrict__ bias,
    float* __restrict__ out) {
  __shared__ __align__(16) float sW[16 * C_];   // W tile, 16KB
  __shared__ int   sOff[4][16];                 // corner pixel index per n
  __shared__ float sWgt[4][16];                 // corner weight (w * valid)
  __shared__ float sStage[8 * 256];             // per-wave 16x16 D staging

  const int tid = threadIdx.x;
  const int n0  = blockIdx.x * 16;

  if (tid < 16) {
    const int n = n0 + tid;
    const float gx = fminf(fmaxf(mu[2 * n + 0], -1.f), 1.f);
    const float gy = fminf(fmaxf(mu[2 * n + 1], -1.f), 1.f);
    const float ix = ((gx + 1.f) * (float)WD_ - 1.f) * 0.5f;
    const float iy = ((gy + 1.f) * (float)H_ - 1.f) * 0.5f;
    const float x0f = floorf(ix), y0f = floorf(iy);
    const int   x0 = (int)x0f, y0 = (int)y0f;
    const float wx1 = ix - x0f, wx0 = 1.f - wx1;
    const float wy1 = iy - y0f, wy0 = 1.f - wy1;
    const int   xs[2]  = {x0, x0 + 1};
    const int   ys[2]  = {y0, y0 + 1};
    const float wxs[2] = {wx0, wx1};
    const float wys[2] = {wy0, wy1};
#pragma unroll
    for (int ky = 0; ky < 2; ++ky) {
#pragma unroll
      for (int kx = 0; kx < 2; ++kx) {
        const int k  = ky * 2 + kx;
        const int xi = xs[kx], yi = ys[ky];
        const bool valid = (xi >= 0) && (xi < WD_) && (yi >= 0) && (yi < H_);
        const int xc = min(max(xi, 0), WD_ - 1);
        const int yc = min(max(yi, 0), H_ - 1);
        sOff[k][tid] = yc * WD_ + xc;
        sWgt[k][tid] = wxs[kx] * wys[ky] * (valid ? 1.f : 0.f);
      }
    }
  }
  {  // stage W tile: rows n0..n0+15 are one contiguous 16KB block
    const float4* src = (const float4*)(Wm + (size_t)n0 * C_);
    float4*       dst = (float4*)sW;
    for (int i = tid; i < (16 * C_) / 4; i += 256) dst[i] = src[i];
  }
  __syncthreads();

  const int wave = tid >> 5;
  const int lane = tid & 31;
  const int nl   = lane & 15;   // n row/col this lane serves
  const int half = lane >> 4;   // selects K pair {0,1} vs {2,3}

  long  poff[4];
  float pw[4];
#pragma unroll
  for (int k = 0; k < 4; ++k) {
    poff[k] = (long)sOff[k][nl] * (long)PS;
    pw[k]   = sWgt[k][nl];
  }
  const float* wrow = sW + nl * C_;
  float*       stg  = sStage + wave * 256;

#pragma unroll
  for (int bp = 0; bp < 2; ++bp) {
    const int biA = wave + bp * 16;     // batches biA and biA+8 this pass
    const int biB = biA + 8;
    const float* xbA = xsrc + (size_t)biA * (size_t)(P_ * C_);
    const float* xbB = xsrc + (size_t)biB * (size_t)(P_ * C_);

    if (bp == 0 && CS == 1) {
      // warm next pass's gather rows in the WGP-level cache
      // (global_prefetch_b8; rows are 1KB, each lane covers part of its row)
      const float* nA = xsrc + (size_t)(wave + 16) * (size_t)(P_ * C_);
      const float* nB = xsrc + (size_t)(wave + 24) * (size_t)(P_ * C_);
#pragma unroll
      for (int k = 0; k < 4; ++k) {
        const long o = poff[k] + (long)half * 128;
        __builtin_prefetch(nA + o, 0, 2);
        __builtin_prefetch(nA + o + 64, 0, 2);
        __builtin_prefetch(nB + o, 0, 2);
        __builtin_prefetch(nB + o + 64, 0, 2);
      }
    }

    v8f acc0 = {0.f, 0.f, 0.f, 0.f, 0.f, 0.f, 0.f, 0.f};
    v8f acc1 = {0.f, 0.f, 0.f, 0.f, 0.f, 0.f, 0.f, 0.f};
#pragma unroll 4
    for (int c0 = 0; c0 < C_; c0 += 4) {
      const int c = c0 + 2 * half;
      const v2f a = *(const v2f*)(wrow + c);   // shared A: W[nl, c..c+1]

      // issue all 8 corner loads first, then combine (batched waits)
      v2f gA[4], gB[4];
#pragma unroll
      for (int k = 0; k < 4; ++k) {
        if constexpr (CS == 1) {
          gA[k] = *(const v2f*)(xbA + poff[k] + c);
          gB[k] = *(const v2f*)(xbB + poff[k] + c);
        } else {
          const float* sA = xbA + poff[k] + (long)c * (long)CS;
          const float* sB = xbB + poff[k] + (long)c * (long)CS;
          gA[k].x = sA[0]; gA[k].y = sA[CS];
          gB[k].x = sB[0]; gB[k].y = sB[CS];
        }
      }
      float a0 = pw[0] * gA[0].x, a1 = pw[0] * gA[0].y;
      float b0 = pw[0] * gB[0].x, b1 = pw[0] * gB[0].y;
#pragma unroll
      for (int k = 1; k < 4; ++k) {
        a0 = fmaf(pw[k], gA[k].x, a0); a1 = fmaf(pw[k], gA[k].y, a1);
        b0 = fmaf(pw[k], gB[k].x, b0); b1 = fmaf(pw[k], gB[k].y, b1);
      }
      const v2f bvA = {a0, a1};
      const v2f bvB = {b0, b1};
      acc0 = __builtin_amdgcn_wmma_f32_16x16x4_f32(false, a, false, bvA,
                                                   (short)0, acc0, false, false);
      acc1 = __builtin_amdgcn_wmma_f32_16x16x4_f32(false, a, false, bvB,
                                                   (short)0, acc1, false, false);
    }

    // stage D matrices to LDS, pull diagonals z[bi, n0+n]
#pragma unroll
    for (int r = 0; r < 8; ++r) stg[(r + 8 * half) * 16 + nl] = acc0[r];
    __syncthreads();
    if (lane < 16) {
      const float z = stg[lane * 16 + lane] + bias[n0 + lane];
      const float e = (z > 0.f) ? z : (__expf(z) - 1.f);
      out[(size_t)biA * N_ + (n0 + lane)] = e + 1.f;
    }
    __syncthreads();
#pragma unroll
    for (int r = 0; r < 8; ++r) stg[(r + 8 * half) * 16 + nl] = acc1[r];
    __syncthreads();
    if (lane < 16) {
      const float z = stg[lane * 16 + lane] + bias[n0 + lane];
      const float e = (z > 0.f) ? z : (__expf(z) - 1.f);
      out[(size_t)biB * N_ + (n0 + lane)] = e + 1.f;
    }
    __syncthreads();
  }
}

extern "C" void kernel_launch(void* const* d_in, const int* in_sizes, int n_in,
                              void* d_out, int out_size, void* d_ws,
                              size_t ws_size, hipStream_t stream) {
  const float* x  = (const float*)d_in[0];
  const float* mu = (const float*)d_in[1];
  const float* Wm = (const float*)d_in[2];
  const float* bv = (const float*)d_in[3];
  float*       out = (float*)d_out;

  const size_t xbytes = (size_t)B_ * C_ * P_ * sizeof(float);
  if (ws_size >= xbytes) {
    float* xt = (float*)d_ws;
    dim3 g(P_ / 32, C_ / 32, B_);
    transpose_x<<<g, 256, 0, stream>>>(x, xt);
    // transposed layout: pixel stride = C_, channel stride = 1 (coalesced)
    readout_wmma<C_, 1><<<N_ / 16, 256, 0, stream>>>(xt, mu, Wm, bv, out);
  } else {
    // fallback: gather straight from (B,C,H,W)
    readout_wmma<1, P_><<<N_ / 16, 256, 0, stream>>>(x, mu, Wm, bv, out);
  }
}